// FLAME_27900107555504
// MI455X (gfx1250) — compile-verified
//
#include <hip/hip_runtime.h>
#include <math.h>

// Problem sizes (from reference)
#define BATCH 2048
#define NV    5023
#define NJ    5
#define NSH   100
#define NEX   50
#define LBETA 150          // NSH+NEX
#define NPB   36           // 9*(NJ-1)
#define KTOT  192          // padded K: 150 betas + 36 pose feature + 6 zeros
#define NCOL  15069        // NV*3
#define NPAD  15072        // 314 * 48 (column padding for exact tiling)
#define NVT   314          // vertex tiles (48 columns = 16 vertices each)
#define NBT   128          // batch tiles (16 rows each)

typedef float v2f __attribute__((ext_vector_type(2)));
typedef float v8f __attribute__((ext_vector_type(8)));

// Workspace layout (in floats)
#define OFF_DIRS  0u
#define OFF_FEAT  (KTOT * NPAD)                       // 2,893,824
#define OFF_RELT  (OFF_FEAT + BATCH * KTOT)           // +393,216
#define OFF_JDIR  (OFF_RELT + BATCH * NJ * 12)        // +122,880
#define OFF_JTMP  (OFF_JDIR + NJ * 3 * LBETA)         // +2,250

// ---------------------------------------------------------------------------
// Kernel 1: build combined direction matrix Dirs[192][15072]
//   rows 0..149   : shapedirs transposed  (Dirs[l][v*3+k] = shapedirs[v,k,l])
//   rows 150..185 : posedirs              (row-major (36, 15069))
//   rows 186..191 : zero ; cols >= 15069 : zero
// ---------------------------------------------------------------------------
__global__ void build_dirs(const float* __restrict__ sdirs,
                           const float* __restrict__ pdirs,
                           float* __restrict__ dirs) {
  int idx = blockIdx.x * blockDim.x + threadIdx.x;
  if (idx >= KTOT * NPAD) return;
  int r = idx / NPAD;
  int c = idx % NPAD;
  float val = 0.0f;
  if (c < NCOL) {
    if (r < LBETA) {
      int v = c / 3, k = c % 3;
      val = sdirs[(size_t)v * (3 * LBETA) + k * LBETA + r];
    } else if (r < LBETA + NPB) {
      val = pdirs[(size_t)(r - LBETA) * NCOL + c];
    }
  }
  dirs[idx] = val;
}

// ---------------------------------------------------------------------------
// Kernel 2: Jdirs[(j*3+k)][l] = sum_v Jreg[j,v]*shapedirs[v,k,l]
//           Jtmp[j*3+k]       = sum_v Jreg[j,v]*v_template[v,k]
// 15 blocks (one per (j,k)), threads over l.
// ---------------------------------------------------------------------------
__global__ void build_jdirs(const float* __restrict__ jreg,
                            const float* __restrict__ sdirs,
                            const float* __restrict__ vtemp,
                            float* __restrict__ jdirs,
                            float* __restrict__ jtmp) {
  int jk = blockIdx.x;           // 0..14
  int j = jk / 3, k = jk % 3;
  int t = threadIdx.x;
  if (t < LBETA) {
    float s = 0.0f;
    for (int v = 0; v < NV; ++v)
      s += jreg[(size_t)j * NV + v] * sdirs[(size_t)v * (3 * LBETA) + k * LBETA + t];
    jdirs[jk * LBETA + t] = s;
  } else if (t == LBETA) {
    float s = 0.0f;
    for (int v = 0; v < NV; ++v)
      s += jreg[(size_t)j * NV + v] * vtemp[v * 3 + k];
    jtmp[jk] = s;
  }
}

// ---------------------------------------------------------------------------
// Kernel 3: per-batch small work (one thread per batch):
//   betas -> feat[0:150], Rodrigues, pose_feature -> feat[150:186], feat[186:192]=0,
//   Jts, kinematic chain, posed_joints (output), rel_transforms (workspace).
// ---------------------------------------------------------------------------
__global__ void batch_kernel(const float* __restrict__ shape,
                             const float* __restrict__ expr,
                             const float* __restrict__ pose,
                             const float* __restrict__ jdirs,
                             const float* __restrict__ jtmp,
                             float* __restrict__ feat,
                             float* __restrict__ relT,
                             float* __restrict__ joints) {
  int b = blockIdx.x * blockDim.x + threadIdx.x;
  if (b >= BATCH) return;

  float beta[LBETA];
  for (int l = 0; l < NSH; ++l) beta[l] = shape[(size_t)b * NSH + l];
  for (int l = 0; l < NEX; ++l) beta[NSH + l] = expr[(size_t)b * NEX + l];

  float* f = feat + (size_t)b * KTOT;
  for (int l = 0; l < LBETA; ++l) f[l] = beta[l];
  for (int l = LBETA + NPB; l < KTOT; ++l) f[l] = 0.0f;

  // Rodrigues (matches reference incl. +1e-8 inside the norm)
  float R[NJ][9];
  for (int j = 0; j < NJ; ++j) {
    float rx = pose[(size_t)b * (NJ * 3) + j * 3 + 0];
    float ry = pose[(size_t)b * (NJ * 3) + j * 3 + 1];
    float rz = pose[(size_t)b * (NJ * 3) + j * 3 + 2];
    float ax = rx + 1e-8f, ay = ry + 1e-8f, az = rz + 1e-8f;
    float ang = sqrtf(ax * ax + ay * ay + az * az);
    float inv = 1.0f / ang;
    float dx = rx * inv, dy = ry * inv, dz = rz * inv;
    float sn = sinf(ang), cs = cosf(ang);
    float K[9] = {0.f, -dz, dy, dz, 0.f, -dx, -dy, dx, 0.f};
    float K2[9];
    for (int r = 0; r < 3; ++r)
      for (int c = 0; c < 3; ++c)
        K2[r * 3 + c] = K[r * 3 + 0] * K[0 + c] + K[r * 3 + 1] * K[3 + c] + K[r * 3 + 2] * K[6 + c];
    for (int e = 0; e < 9; ++e) {
      float id = (e % 4 == 0) ? 1.0f : 0.0f;
      R[j][e] = id + sn * K[e] + (1.0f - cs) * K2[e];
    }
  }

  // pose_feature: joints 1..4, (R - I) row-major
  for (int j = 1; j < NJ; ++j)
    for (int e = 0; e < 9; ++e) {
      float id = (e % 4 == 0) ? 1.0f : 0.0f;
      f[LBETA + (j - 1) * 9 + e] = R[j][e] - id;
    }

  // Joint locations: Jt = Jtmp + Jdirs . beta
  float Jt[NJ][3];
  for (int j = 0; j < NJ; ++j)
    for (int k = 0; k < 3; ++k) {
      float s = jtmp[j * 3 + k];
      const float* jd = jdirs + (j * 3 + k) * LBETA;
      for (int l = 0; l < LBETA; ++l) s += jd[l] * beta[l];
      Jt[j][k] = s;
    }

  const int par[NJ] = {0, 0, 1, 1, 1};

  // Local affine transforms (3x4, implicit bottom row [0 0 0 1])
  float A[NJ][12], Tw[NJ][12];
  for (int j = 0; j < NJ; ++j) {
    float tx = Jt[j][0], ty = Jt[j][1], tz = Jt[j][2];
    if (j > 0) { tx -= Jt[par[j]][0]; ty -= Jt[par[j]][1]; tz -= Jt[par[j]][2]; }
    for (int r = 0; r < 3; ++r) {
      A[j][r * 4 + 0] = R[j][r * 3 + 0];
      A[j][r * 4 + 1] = R[j][r * 3 + 1];
      A[j][r * 4 + 2] = R[j][r * 3 + 2];
    }
    A[j][0 * 4 + 3] = tx; A[j][1 * 4 + 3] = ty; A[j][2 * 4 + 3] = tz;
  }
  for (int e = 0; e < 12; ++e) Tw[0][e] = A[0][e];
  for (int i = 1; i < NJ; ++i) {
    int p = par[i];
    for (int r = 0; r < 3; ++r)
      for (int c = 0; c < 4; ++c) {
        float s = Tw[p][r * 4 + 0] * A[i][0 * 4 + c]
                + Tw[p][r * 4 + 1] * A[i][1 * 4 + c]
                + Tw[p][r * 4 + 2] * A[i][2 * 4 + c];
        if (c == 3) s += Tw[p][r * 4 + 3];
        Tw[i][r * 4 + c] = s;
      }
  }

  // posed joints (second output) + relative transforms (workspace)
  for (int j = 0; j < NJ; ++j) {
    for (int k = 0; k < 3; ++k)
      joints[(size_t)b * (NJ * 3) + j * 3 + k] = Tw[j][k * 4 + 3];
    float* rt = relT + (size_t)b * (NJ * 12) + j * 12;
    for (int r = 0; r < 3; ++r) {
      float tj = Tw[j][r * 4 + 0] * Jt[j][0] + Tw[j][r * 4 + 1] * Jt[j][1]
               + Tw[j][r * 4 + 2] * Jt[j][2];
      rt[r * 4 + 0] = Tw[j][r * 4 + 0];
      rt[r * 4 + 1] = Tw[j][r * 4 + 1];
      rt[r * 4 + 2] = Tw[j][r * 4 + 2];
      rt[r * 4 + 3] = Tw[j][r * 4 + 3] - tj;
    }
  }
}

// ---------------------------------------------------------------------------
// Kernel 4: fused GEMM (K=192, f32 WMMA 16x16x4) + LBS skinning.
// One wave32 per (16-batch x 48-column) tile; 48 cols = 16 vertices * xyz.
// A layout (ISA 7.12.2, 32-bit A 16x4): v0 = K0/K2 (lane halves), v1 = K1/K3.
// B layout assumed symmetric (col = lane&15, K across vgpr+half).
// C layout: vgpr r -> M = r + 8*(lane>=16), N = lane&15.
// ---------------------------------------------------------------------------
__global__ __launch_bounds__(32)
void fused_gemm_skin(const float* __restrict__ dirs,
                     const float* __restrict__ feat,
                     const float* __restrict__ relT,
                     const float* __restrict__ lbs,
                     const float* __restrict__ vtemp,
                     float* __restrict__ verts) {
  const int vt = blockIdx.x;          // vertex tile
  const int bt = blockIdx.y;          // batch tile
  const int lane = threadIdx.x;       // 0..31
  const int half = lane >> 4;
  const int l15 = lane & 15;

  __shared__ float vp[16 * 48];       // v_posed tile [m][n]
  __shared__ float rts[16 * 60];      // rel transforms for 16 batches
  __shared__ float lws[16 * 5];       // lbs weights for 16 vertices

  v8f c0 = {}; v8f c1 = {}; v8f c2 = {};
  const float* fa = feat + (size_t)(bt * 16 + l15) * KTOT;
  const int col0 = vt * 48 + l15;

  for (int k0 = 0; k0 < KTOT; k0 += 4) {
    v2f a;
    a.x = fa[k0 + half * 2 + 0];
    a.y = fa[k0 + half * 2 + 1];
    const float* dr0 = dirs + (size_t)(k0 + half * 2 + 0) * NPAD + col0;
    const float* dr1 = dirs + (size_t)(k0 + half * 2 + 1) * NPAD + col0;
    v2f b0; b0.x = dr0[0];  b0.y = dr1[0];
    v2f b1; b1.x = dr0[16]; b1.y = dr1[16];
    v2f b2; b2.x = dr0[32]; b2.y = dr1[32];
    c0 = __builtin_amdgcn_wmma_f32_16x16x4_f32(false, a, false, b0, (short)0, c0, false, false);
    c1 = __builtin_amdgcn_wmma_f32_16x16x4_f32(false, a, false, b1, (short)0, c1, false, false);
    c2 = __builtin_amdgcn_wmma_f32_16x16x4_f32(false, a, false, b2, (short)0, c2, false, false);
  }

  // v_template for this lane's three columns
  float tv[3];
#pragma unroll
  for (int t = 0; t < 3; ++t) {
    int col = vt * 48 + t * 16 + l15;
    int v = col / 3, k = col % 3;
    tv[t] = (v < NV) ? vtemp[v * 3 + k] : 0.0f;
  }

  // Stage v_posed tile to LDS
#pragma unroll
  for (int r = 0; r < 8; ++r) {
    int m = r + half * 8;
    vp[m * 48 + 0 * 16 + l15] = c0[r] + tv[0];
    vp[m * 48 + 1 * 16 + l15] = c1[r] + tv[1];
    vp[m * 48 + 2 * 16 + l15] = c2[r] + tv[2];
  }

  // Load rel-transform + lbs-weight tiles
  for (int i = lane; i < 16 * 60; i += 32)
    rts[i] = relT[(size_t)(bt * 16 + i / 60) * 60 + (i % 60)];
  for (int i = lane; i < 16 * 5; i += 32) {
    int v = vt * 16 + i / 5;
    lws[i] = (v < NV) ? lbs[(size_t)v * NJ + (i % 5)] : 0.0f;
  }
  __syncthreads();

  // Skinning: 256 (batch, vertex) pairs / 32 lanes
  for (int p = lane; p < 256; p += 32) {
    int m = p & 15;
    int vl = p >> 4;
    int v = vt * 16 + vl;
    if (v >= NV) continue;
    float acc[12];
#pragma unroll
    for (int e = 0; e < 12; ++e) acc[e] = 0.0f;
#pragma unroll
    for (int j = 0; j < NJ; ++j) {
      float w = lws[vl * 5 + j];
#pragma unroll
      for (int e = 0; e < 12; ++e) acc[e] += w * rts[m * 60 + j * 12 + e];
    }
    float x = vp[m * 48 + vl * 3 + 0];
    float y = vp[m * 48 + vl * 3 + 1];
    float z = vp[m * 48 + vl * 3 + 2];
    size_t base = ((size_t)(bt * 16 + m) * NV + v) * 3;
    verts[base + 0] = acc[0] * x + acc[1] * y + acc[2]  * z + acc[3];
    verts[base + 1] = acc[4] * x + acc[5] * y + acc[6]  * z + acc[7];
    verts[base + 2] = acc[8] * x + acc[9] * y + acc[10] * z + acc[11];
  }
}

// ---------------------------------------------------------------------------
extern "C" void kernel_launch(void* const* d_in, const int* in_sizes, int n_in,
                              void* d_out, int out_size, void* d_ws, size_t ws_size,
                              hipStream_t stream) {
  const float* shape = (const float*)d_in[0];
  const float* expr  = (const float*)d_in[1];
  const float* pose  = (const float*)d_in[2];
  const float* vtemp = (const float*)d_in[3];
  const float* sdirs = (const float*)d_in[4];
  const float* pdirs = (const float*)d_in[5];
  const float* jreg  = (const float*)d_in[6];
  const float* lbs   = (const float*)d_in[7];
  (void)in_sizes; (void)n_in; (void)out_size; (void)ws_size;

  float* out    = (float*)d_out;
  float* verts  = out;
  float* joints = out + (size_t)BATCH * NV * 3;

  float* ws    = (float*)d_ws;
  float* dirs  = ws + OFF_DIRS;
  float* feat  = ws + OFF_FEAT;
  float* relT  = ws + OFF_RELT;
  float* jdirs = ws + OFF_JDIR;
  float* jtmp  = ws + OFF_JTMP;

  build_dirs<<<(KTOT * NPAD + 255) / 256, 256, 0, stream>>>(sdirs, pdirs, dirs);
  build_jdirs<<<NJ * 3, 256, 0, stream>>>(jreg, sdirs, vtemp, jdirs, jtmp);
  batch_kernel<<<(BATCH + 255) / 256, 256, 0, stream>>>(shape, expr, pose, jdirs, jtmp,
                                                        feat, relT, joints);
  fused_gemm_skin<<<dim3(NVT, NBT), 32, 0, stream>>>(dirs, feat, relT, lbs, vtemp, verts);
}